// AsynBaseStem_13511967113612
// MI455X (gfx1250) — compile-verified
//
#include <hip/hip_runtime.h>
#include <stdint.h>

// CDNA5 (gfx1250) wave32 WMMA types
typedef __attribute__((ext_vector_type(2))) float v2f;   // A/B operand of V_WMMA_F32_16X16X4_F32
typedef __attribute__((ext_vector_type(8))) float v8f;   // 16x16 f32 C/D accumulator

#define HDIM   640
#define WDIM   640
#define CIN    3
#define COUT   64
#define PADV   3
#define NSITES 100000
#define KDIM   147      // 7*7*3
#define KPAD   148      // A rows padded; element 147 zeroed
#define NKP    74       // K pairs (148/2)
#define BSTRIDE 160     // Bs2 kp-row stride in dwords: 128 data + 32 pad (== 32 mod 64 banks)
#define BSZ    (NKP * BSTRIDE)          // 11,840 dwords (== 0 mod 64) at LDS offset 0
#define ASZ    (8 * 16 * KPAD)          // 18,944 dwords
#define POUT   319      // floor((640-3)/2)+1
#define BN_EPS 1e-5f

// ---------------------------------------------------------------------------
// Zero the pooled output (26 MB) with wide stores.
// ---------------------------------------------------------------------------
__global__ void zero_out_kernel(float4* __restrict__ out, int n4) {
  int i = blockIdx.x * blockDim.x + threadIdx.x;
  if (i < n4) out[i] = make_float4(0.f, 0.f, 0.f, 0.f);
}

// ---------------------------------------------------------------------------
// Fused: LDS-staged implicit GEMM (f32 WMMA) + BN + ReLU + atomic-max scatter.
//
// Per 256-thread block: 8 waves x 16 sites = 128 sites.
//  - Single LDS buffer, explicitly partitioned: B at offset 0 so EVERY
//    ds_load_b64 in the K-loop addresses as base + 16-bit immediate
//    (B spans < 64 KB), eliminating per-load address VALU.
//  - B staged in WMMA-operand order: Bs2[kp][n*2 + q] = W[2kp+q][n], kp-row
//    stride 160 dwords -> each lane's pair is one conflict-free ds_load_b64.
//  - A staged per wave as [site][148] (stride 148 == 20 mod 64 banks ->
//    conflict-free), lane pair fetch is one ds_load_b64.
//  - K-loop: FULLY UNROLLED 37 steps, 4 x V_WMMA_F32_16X16X4_F32 per step
//    (fixed accumulator registers, in-place accumulate).
//  - Epilogue: BN+ReLU then atomic u32-max into the <=2x2 pooled cells each
//    site overlaps (valid: feats >= 0, background 0, and u32 bit-pattern
//    order == f32 order for non-negative values).
// ---------------------------------------------------------------------------
__global__ __launch_bounds__(256)
void stem_wmma_kernel(const int*   __restrict__ loc,     // [NSITES,2]
                      const float* __restrict__ fm,      // [640,640,3]
                      const float* __restrict__ wgt,     // [7,7,3,64] == [147,64]
                      const float* __restrict__ gamma,
                      const float* __restrict__ beta,
                      const float* __restrict__ rmean,
                      const float* __restrict__ rvar,
                      float*       __restrict__ out)     // [319,319,64]
{
  __shared__ float lds[BSZ + ASZ];      // 123,136 B total
  float* Bs2 = lds;                     // [NKP][BSTRIDE] at LDS offset 0

  // ---- Block-wide weight stage into WMMA operand order ----
  for (int idx = threadIdx.x; idx < NKP * COUT; idx += 256) {
    const int kp = idx >> 6;            // /64
    const int n  = idx & 63;
    const int k0 = kp * 2;
    v2f w;
    w[0] = wgt[k0 * COUT + n];                                  // k0 <= 146
    w[1] = (k0 + 1 < KDIM) ? wgt[(k0 + 1) * COUT + n] : 0.f;    // zero pad 147
    *(v2f*)&Bs2[kp * BSTRIDE + n * 2] = w;
  }

  const int lane = threadIdx.x & 31;
  const int wave = threadIdx.x >> 5;
  const int half = lane >> 4;
  const int nl   = lane & 15;
  const int base = blockIdx.x * 128 + wave * 16;

  float* waveA = lds + BSZ + wave * (16 * KPAD);   // this wave's A tile

  // ---- Per-wave patch stage: lane pair (nl, nl+16) stages site base+nl ----
  // half==0 -> patch rows i = 0..2, half==1 -> rows i = 3..6 (+ k=147 pad).
  {
    const int  m      = base + nl;
    const bool siteOk = (m < NSITES);
    const int  rA = siteOk ? loc[2 * m]     : 0;
    const int  cA = siteOk ? loc[2 * m + 1] : 0;
    float* myA = waveA + nl * KPAD;

    #pragma unroll 1
    for (int ii = 0; ii < 4; ++ii) {
      const bool rowActive = (half == 1) || (ii < 3);
      int i = ii + 3 * half;            // half0: 0..3 (3 masked), half1: 3..6
      if (i > 6) i = 6;                 // harmless clamp for the masked row
      const int  rr  = rA - PADV + i;
      const bool rok = siteOk && (rr >= 0) && (rr < HDIM) && rowActive;
      const int  rrc = min(max(rr, 0), HDIM - 1);    // always-in-bounds row
      const int  kb  = i * 21;
      #pragma unroll
      for (int j = 0; j < 7; ++j) {
        const int  cc  = cA - PADV + j;
        const bool ok  = rok && (cc >= 0) && (cc < WDIM);
        const int  ccc = min(max(cc, 0), WDIM - 1);  // always-in-bounds col
        const float* p = fm + (rrc * WDIM + ccc) * CIN;
        #pragma unroll
        for (int c = 0; c < 3; ++c) {
          const float v = p[c];                      // clamped => safe load
          if (rowActive) myA[kb + j * 3 + c] = ok ? v : 0.f;
        }
      }
    }
    if (half == 1) myA[KDIM] = 0.f;     // k = 147 pad
  }

  __syncthreads();

  // ---- K-loop (fully unrolled): D = A(16x148) x B(148x64) ----
  const float* aPtr = waveA + nl * KPAD + 2 * half;        // 8B aligned
  const float* bPtr = &Bs2[half * BSTRIDE + nl * 2];       // 8B aligned; all
                                                           // step offsets fit
                                                           // the 16-bit DS imm
  v8f acc[4];
  #pragma unroll
  for (int t = 0; t < 4; ++t) acc[t] = (v8f){};

  #pragma unroll
  for (int s = 0; s < 37; ++s) {
    const v2f a  = *(const v2f*)(aPtr + 4 * s);
    const float* br = bPtr + 2 * BSTRIDE * s;  // byte offset 1280*s <= 46,080
    const v2f b0 = *(const v2f*)(br +  0);     // N-tile 0 (cols  0..15)
    const v2f b1 = *(const v2f*)(br + 32);     // N-tile 1 (cols 16..31)
    const v2f b2 = *(const v2f*)(br + 64);     // N-tile 2 (cols 32..47)
    const v2f b3 = *(const v2f*)(br + 96);     // N-tile 3 (cols 48..63)

    acc[0] = __builtin_amdgcn_wmma_f32_16x16x4_f32(false, a, false, b0, (short)0, acc[0], false, false);
    acc[1] = __builtin_amdgcn_wmma_f32_16x16x4_f32(false, a, false, b1, (short)0, acc[1], false, false);
    acc[2] = __builtin_amdgcn_wmma_f32_16x16x4_f32(false, a, false, b2, (short)0, acc[2], false, false);
    acc[3] = __builtin_amdgcn_wmma_f32_16x16x4_f32(false, a, false, b3, (short)0, acc[3], false, false);
  }

  // ---- BN params once per channel tile ----
  float inv[4], bb[4];
  #pragma unroll
  for (int t = 0; t < 4; ++t) {
    const int n = t * 16 + nl;
    inv[t] = gamma[n] * rsqrtf(rvar[n] + BN_EPS);
    bb[t]  = beta[n] - rmean[n] * inv[t];
  }

  // ---- BN + ReLU + pooled atomic-max scatter ----
  #pragma unroll
  for (int r = 0; r < 8; ++r) {
    const int m = base + r + 8 * half;  // C/D row M = r + 8*(lane>>4)
    if (m >= NSITES) continue;
    const int rp = loc[2 * m];
    const int cp = loc[2 * m + 1];
    const int prl = max(0, (rp - 1) >> 1);        // ceil((rp-2)/2)
    const int prh = min(POUT - 1, rp >> 1);
    const int pcl = max(0, (cp - 1) >> 1);
    const int pch = min(POUT - 1, cp >> 1);
    #pragma unroll
    for (int t = 0; t < 4; ++t) {
      const float val = fmaxf(acc[t][r] * inv[t] + bb[t], 0.f);
      if (val <= 0.f) continue;       // 0 never beats the zero-initialized out
      const unsigned uval = __float_as_uint(val);
      const int n = t * 16 + nl;
      for (int pr = prl; pr <= prh; ++pr)
        for (int pc = pcl; pc <= pch; ++pc)
          atomicMax((unsigned int*)&out[(pr * POUT + pc) * COUT + n], uval);
    }
  }
}

// ---------------------------------------------------------------------------
extern "C" void kernel_launch(void* const* d_in, const int* in_sizes, int n_in,
                              void* d_out, int out_size, void* d_ws, size_t ws_size,
                              hipStream_t stream) {
  const int*   loc   = (const int*)  d_in[0];  // update_location [N,2]
  const float* fm    = (const float*)d_in[1];  // feature_map [640,640,3]
  const float* wgt   = (const float*)d_in[2];  // weight [7,7,3,64]
  const float* gamma = (const float*)d_in[3];
  const float* beta  = (const float*)d_in[4];
  const float* rmean = (const float*)d_in[5];
  const float* rvar  = (const float*)d_in[6];
  float*       out   = (float*)d_out;          // [319,319,64]

  const int n4 = out_size / 4;                 // 319*319*64 divisible by 4
  zero_out_kernel<<<(n4 + 255) / 256, 256, 0, stream>>>((float4*)d_out, n4);

  const int nblocks = (NSITES + 127) / 128;    // 128 sites per 256-thread block
  stem_wmma_kernel<<<nblocks, 256, 0, stream>>>(loc, fm, wgt, gamma, beta,
                                                rmean, rvar, out);
}